// Gate_28922309771625
// MI455X (gfx1250) — compile-verified
//
#include <hip/hip_runtime.h>
#include <hip/hip_bf16.h>
#include <stdint.h>

// MoE gate (DeepSeek-style group-limited routing) for MI455X / gfx1250.
// GEMM: logits[T,64] = x[T,2048] @ W[64,2048]^T  via V_WMMA_F32_16X16X4_F32
// (fp32 matrix path: exact vs reference, and we are HBM-bound on x anyway).

typedef __attribute__((ext_vector_type(2))) float v2f;
typedef __attribute__((ext_vector_type(8))) float v8f;

#define DIM            2048
#define N_EXPERTS      64
#define N_GROUPS       8
#define GROUP_SIZE     8
#define TOPK_GROUPS    4
#define TOPK           8
#define ROUTE_SCALE    1.0f
#define TOK_PER_BLOCK  128
#define THREADS        256
#define NEG_INF        (-3.402823466e+38f)

__global__ __launch_bounds__(THREADS)
void moe_gate_kernel(const float* __restrict__ x,
                     const float* __restrict__ w,
                     const float* __restrict__ bias,
                     float* __restrict__ outW,
                     int*   __restrict__ outI,
                     int T)
{
    // logits staging: 128 tokens x 64 experts x f32 = 32 KB LDS
    __shared__ float lds[TOK_PER_BLOCK * N_EXPERTS];

    const int lane  = threadIdx.x & 31;
    const int wave  = threadIdx.x >> 5;
    const int row   = lane & 15;          // M (token) for A, N (expert) for B
    const int khalf = (lane >> 4) << 1;   // lanes 0-15 -> K+{0,1}, lanes 16-31 -> K+{2,3}
    const int tok0  = blockIdx.x * TOK_PER_BLOCK;

    // clamp token row so partial tail blocks never read OOB (results discarded)
    int tokRow = tok0 + wave * 16 + row;
    if (tokRow >= T) tokRow = T - 1;

    const float* xp  = x + (size_t)tokRow * DIM + khalf;
    const float* wp0 = w + (size_t)(row +  0) * DIM + khalf;
    const float* wp1 = w + (size_t)(row + 16) * DIM + khalf;
    const float* wp2 = w + (size_t)(row + 32) * DIM + khalf;
    const float* wp3 = w + (size_t)(row + 48) * DIM + khalf;

    v8f acc0 = {}, acc1 = {}, acc2 = {}, acc3 = {};

    #pragma unroll 4
    for (int k = 0; k < DIM; k += 4) {
        v2f a  = *(const v2f*)(xp  + k);
        v2f b0 = *(const v2f*)(wp0 + k);
        v2f b1 = *(const v2f*)(wp1 + k);
        v2f b2 = *(const v2f*)(wp2 + k);
        v2f b3 = *(const v2f*)(wp3 + k);
        // 8 args: (neg_a, A, neg_b, B, c_mod, C, reuse_a, reuse_b)
        acc0 = __builtin_amdgcn_wmma_f32_16x16x4_f32(false, a, false, b0, (short)0, acc0, false, false);
        acc1 = __builtin_amdgcn_wmma_f32_16x16x4_f32(false, a, false, b1, (short)0, acc1, false, false);
        acc2 = __builtin_amdgcn_wmma_f32_16x16x4_f32(false, a, false, b2, (short)0, acc2, false, false);
        acc3 = __builtin_amdgcn_wmma_f32_16x16x4_f32(false, a, false, b3, (short)0, acc3, false, false);
    }

    // C/D layout: VGPR v -> (lanes 0-15: M=v, N=lane), (lanes 16-31: M=v+8, N=lane-16)
    const int tlBase = wave * 16 + ((lane >> 4) << 3);
    #pragma unroll
    for (int v = 0; v < 8; ++v) {
        float* dst = &lds[(tlBase + v) * N_EXPERTS + row];
        dst[ 0] = acc0[v];
        dst[16] = acc1[v];
        dst[32] = acc2[v];
        dst[48] = acc3[v];
    }

    __syncthreads();

    // ---- per-token softmax + group-limited top-k routing ----
    const int t = threadIdx.x;
    const int tok = tok0 + t;
    if (t < TOK_PER_BLOCK && tok < T) {
        float* s = &lds[t * N_EXPERTS];

        // softmax (in place): probs = exp(l - max) / sum
        float mx = s[0];
        for (int e = 1; e < N_EXPERTS; ++e) mx = fmaxf(mx, s[e]);
        float sum = 0.0f;
        for (int e = 0; e < N_EXPERTS; ++e) {
            float v = __expf(s[e] - mx);
            s[e] = v;
            sum += v;
        }
        const float inv = 1.0f / sum;
        for (int e = 0; e < N_EXPERTS; ++e) s[e] *= inv;

        // group score = sum of top-2 of (prob + bias) within each 8-expert group
        float gs[N_GROUPS];
        for (int g = 0; g < N_GROUPS; ++g) {
            float t1 = NEG_INF, t2 = NEG_INF;
            const int base = g * GROUP_SIZE;
            for (int j = 0; j < GROUP_SIZE; ++j) {
                float v = s[base + j] + bias[base + j];
                if (v > t1) { t2 = t1; t1 = v; }
                else if (v > t2) { t2 = v; }
            }
            gs[g] = t1 + t2;
        }

        // top-4 groups (strict > => lowest index wins ties, like jax top_k)
        unsigned gmask = 0;
        for (int i = 0; i < TOPK_GROUPS; ++i) {
            int best = 0; float bv = NEG_INF;
            for (int g = 0; g < N_GROUPS; ++g) {
                if ((gmask >> g) & 1u) continue;
                if (gs[g] > bv) { bv = gs[g]; best = g; }
            }
            gmask |= 1u << best;
        }

        // top-8 experts over masked scores (masked-out groups contribute 0.0,
        // exactly as reference: scores_g * mask)
        uint64_t picked = 0;
        for (int i = 0; i < TOPK; ++i) {
            int best = 0; float bv = NEG_INF;
            for (int e = 0; e < N_EXPERTS; ++e) {
                if ((picked >> e) & 1ull) continue;
                float v = ((gmask >> (e >> 3)) & 1u) ? (s[e] + bias[e]) : 0.0f;
                if (v > bv) { bv = v; best = e; }
            }
            picked |= 1ull << best;
            outW[(size_t)tok * TOPK + i] = s[best] * ROUTE_SCALE; // original softmax prob
            outI[(size_t)tok * TOPK + i] = best;
        }
    }
}

extern "C" void kernel_launch(void* const* d_in, const int* in_sizes, int n_in,
                              void* d_out, int out_size, void* d_ws, size_t ws_size,
                              hipStream_t stream) {
    const float* x    = (const float*)d_in[0];   // [T, 2048]
    const float* w    = (const float*)d_in[1];   // [64, 2048]
    const float* bias = (const float*)d_in[2];   // [64]
    const int T = in_sizes[0] / DIM;

    // d_out = weights f32 [T,8] followed by indices i32 [T,8]
    float* outW = (float*)d_out;
    int*   outI = (int*)d_out + (size_t)T * TOPK;

    const int grid = (T + TOK_PER_BLOCK - 1) / TOK_PER_BLOCK;
    moe_gate_kernel<<<grid, THREADS, 0, stream>>>(x, w, bias, outW, outI, T);
}